// E3nnInteraction_3358664425485
// MI455X (gfx1250) — compile-verified
//
#include <hip/hip_runtime.h>

typedef float v2f __attribute__((ext_vector_type(2)));
typedef float v8f __attribute__((ext_vector_type(8)));

#define N_NODES 20000
#define N_EDGES 200000

// D = A(16x4) * B(4x16) + C, f32, wave32.
// A frag: lane<16 holds row=lane, K={k,k+1}; lane>=16 holds row=lane-16, K={k+2,k+3}
// B frag: lane<16 holds N=lane, K={k,k+1}; lane>=16 holds N=lane-16, K={k+2,k+3}
// C/D:    VGPR r, lane L -> M = r + (L>=16 ? 8 : 0), N = L%16
__device__ __forceinline__ v8f wmma4(v2f a, v2f b, v8f c) {
    return __builtin_amdgcn_wmma_f32_16x16x4_f32(
        /*neg_a=*/false, a, /*neg_b=*/false, b,
        /*c_mod=*/(short)0, c, /*reuse_a=*/false, /*reuse_b=*/false);
}

__device__ __forceinline__ float silu(float x) {
    return x / (1.0f + __expf(-x));
}

// ---------------------------------------------------------------------------
__global__ void zero_kernel(float4* __restrict__ p, int n4) {
    const int stride = gridDim.x * blockDim.x;
    for (int i = blockIdx.x * blockDim.x + threadIdx.x; i < n4; i += stride)
        p[i] = make_float4(0.f, 0.f, 0.f, 0.f);
}

// ---------------------------------------------------------------------------
// Pre-pack a (K x N) row-major weight matrix so a WMMA B-fragment is one
// aligned 8-byte load:  Wp[q*N + n] = { W[2q][n], W[2q+1][n] },  q in [0,K/2)
// In-kernel: lane reads Wp[((k>>2)*2 + half)*N + col]  (v2f element index).
__global__ void pack_b_kernel(const float* __restrict__ W, float2* __restrict__ Wp,
                              int K, int N) {
    const int total  = (K / 2) * N;
    const int stride = gridDim.x * blockDim.x;
    for (int idx = blockIdx.x * blockDim.x + threadIdx.x; idx < total; idx += stride) {
        const int q = idx / N;
        const int n = idx - q * N;
        float2 p;
        p.x = W[(2 * q) * N + n];
        p.y = W[(2 * q + 1) * N + n];
        Wp[idx] = p;
    }
}

// ---------------------------------------------------------------------------
// s_up = s @ W_up0 * inv ; v_up[n,k,m] = sum_u v[n,u,m] W_up1[u,k] * inv
// node_feats row: [0..127]=s, [128 + u*3 + m]=v
// s_up stored (N,128); v_up stored (N,128,3) interleaved
__global__ void __launch_bounds__(256)
node_up_kernel(const float* __restrict__ nf, const v2f* __restrict__ W0p,
               const v2f* __restrict__ W1p, float* __restrict__ s_up,
               float* __restrict__ v_up) {
    const int lane   = threadIdx.x & 31;
    const int wv     = threadIdx.x >> 5;          // 8 waves -> 8 col tiles
    const int lane15 = lane & 15;
    const int half   = lane >> 4;
    const int koff   = half * 2;
    const int mbase  = half * 8;
    const int n0     = blockIdx.x * 16;
    const int col    = wv * 16 + lane15;
    const int ab     = (n0 + lane15) * 512;       // A row base (float units)
    const int ab2    = (n0 + lane15) * 256 + half;// A row base (v2f units)
    const int wbase  = half * 128 + col;          // B base (v2f units)

    const v2f* nf2 = (const v2f*)nf;

    v8f aS = {}, aV0 = {}, aV1 = {}, aV2 = {};
    for (int k = 0; k < 128; k += 4) {
        const int kk = k + koff;
        const int qb = (k >> 2) * 256 + wbase;    // ((k>>2)*2+half)*128 + col
        const v2f b0 = W0p[qb];
        const v2f b1 = W1p[qb];
        const v2f a  = nf2[ab2 + (k >> 1)];
        aS = wmma4(a, b0, aS);
        const int vb = ab + 128 + kk * 3;
        v2f av;
        av.x = nf[vb + 0]; av.y = nf[vb + 3];
        aV0 = wmma4(av, b1, aV0);
        av.x = nf[vb + 1]; av.y = nf[vb + 4];
        aV1 = wmma4(av, b1, aV1);
        av.x = nf[vb + 2]; av.y = nf[vb + 5];
        aV2 = wmma4(av, b1, aV2);
    }

    const float inv = 0.08838834764831845f;  // 1/sqrt(128)
#pragma unroll
    for (int r = 0; r < 8; ++r) {
        const int n = n0 + mbase + r;
        s_up[n * 128 + col] = aS[r] * inv;
        const int vo = (n * 128 + col) * 3;
        v_up[vo + 0] = aV0[r] * inv;
        v_up[vo + 1] = aV1[r] * inv;
        v_up[vo + 2] = aV2[r] * inv;
    }
}

// ---------------------------------------------------------------------------
// One hidden layer of the edge MLP: HDST = silu( (HSRC @ Wp) / 8 ), 16x64.
// k-outer / 4 independent accumulators; A fragment shared across col tiles.
__device__ __forceinline__ void mlp64_layer(const float* HSRC, float* HDST,
                                            const v2f* __restrict__ Wp,
                                            int lane15, int half, int mbase) {
    const v2f* H2    = (const v2f*)HSRC;
    const int abase  = lane15 * 34 + half;        // pitch 68 floats = 34 v2f
    const int wbase  = half * 64 + lane15;
    v8f acc[4] = {{}, {}, {}, {}};
    for (int k = 0; k < 64; k += 4) {
        const v2f a  = H2[abase + (k >> 1)];
        const int qb = (k >> 2) * 128 + wbase;    // ((k>>2)*2+half)*64 + lane15
#pragma unroll
        for (int ct = 0; ct < 4; ++ct) {
            const v2f b = Wp[qb + ct * 16];
            acc[ct] = wmma4(a, b, acc[ct]);
        }
    }
#pragma unroll
    for (int ct = 0; ct < 4; ++ct)
#pragma unroll
        for (int r = 0; r < 8; ++r)
            HDST[(mbase + r) * 68 + ct * 16 + lane15] = silu(acc[ct][r] * 0.125f);
}

// ---------------------------------------------------------------------------
// Fused edge kernel: 4-layer MLP (WMMA, LDS-staged) + tensor-product message
// construction + atomic segment-sum scatter into msg (N,1024).
// One wave handles 16 edges; 4 waves per block.
__global__ void __launch_bounds__(128)
edge_kernel(const float* __restrict__ ea, const float* __restrict__ ef,
            const int* __restrict__ snd, const int* __restrict__ rcv,
            const v2f* __restrict__ w1p, const v2f* __restrict__ w2p,
            const v2f* __restrict__ w3p, const v2f* __restrict__ w4p,
            const float* __restrict__ s_up, const float* __restrict__ v_up,
            float* __restrict__ msg) {
    __shared__ float hA[4][16 * 68];   // 68-float row pitch: bank-conflict free
    __shared__ float hB[4][16 * 68];

    const int lane   = threadIdx.x & 31;
    const int wv     = threadIdx.x >> 5;
    const int lane15 = lane & 15;
    const int half   = lane >> 4;
    const int mbase  = half * 8;
    const int e0     = (blockIdx.x * 4 + wv) * 16;

    float* HA = hA[wv];
    float* HB = hB[wv];

    // warm the packed mlp_w4 (128 KB) into cache: 128 threads x 1 KB apart
    __builtin_prefetch((const float*)w4p + threadIdx.x * 256, 0, 1);

    // ---- per-edge scalars / gather-scatter indices for this lane's rows ----
    int   sidx[8], vidx[8], midx[8];
    float y0a[8], y1xa[8], y1ya[8], y1za[8];
#pragma unroll
    for (int r = 0; r < 8; ++r) {
        const int e    = e0 + mbase + r;
        const int node = snd[e];
        sidx[r] = node * 128;       // into s_up
        vidx[r] = node * 384;       // into v_up
        midx[r] = rcv[e] * 1024;    // into msg
        const float4 y = *(const float4*)(ea + e * 4);
        y0a[r] = y.x; y1xa[r] = y.y; y1ya[r] = y.z; y1za[r] = y.w;
    }

    // ---- layer 1: (16x8) @ mlp_w1(8x64), scale 1/sqrt(8), silu -> HA ----
    {
        const v2f* ef2  = (const v2f*)ef;
        const int eb2   = (e0 + lane15) * 4 + half;
        const int wbase = half * 64 + lane15;
        v8f acc[4] = {{}, {}, {}, {}};
#pragma unroll
        for (int k = 0; k < 8; k += 4) {
            const v2f a  = ef2[eb2 + (k >> 1)];
            const int qb = (k >> 2) * 128 + wbase;
#pragma unroll
            for (int ct = 0; ct < 4; ++ct) {
                const v2f b = w1p[qb + ct * 16];
                acc[ct] = wmma4(a, b, acc[ct]);
            }
        }
#pragma unroll
        for (int ct = 0; ct < 4; ++ct)
#pragma unroll
            for (int r = 0; r < 8; ++r)
                HA[(mbase + r) * 68 + ct * 16 + lane15] = silu(acc[ct][r] * 0.35355339059327373f);
    }
    asm volatile("s_wait_dscnt 0" ::: "memory");

    // ---- layer 2: HA -> HB ; layer 3: HB -> HA ----
    mlp64_layer(HA, HB, w2p, lane15, half, mbase);
    asm volatile("s_wait_dscnt 0" ::: "memory");
    mlp64_layer(HB, HA, w3p, lane15, half, mbase);
    asm volatile("s_wait_dscnt 0" ::: "memory");

    // ---- layer 4 (HA @ mlp_w4(64x512), scale 1/8) fused with messages ----
    const float c  = 0.08838834764831845f;   // 1/sqrt(128)
    const float c3 = 0.05103103630798287f;   // 1/sqrt(3*128)
    const v2f* H2   = (const v2f*)HA;
    const int abase = lane15 * 34 + half;
#pragma unroll 1
    for (int g = 0; g < 8; ++g) {            // 4 col tiles per pass
        const int wbase = half * 512 + g * 64 + lane15;
        v8f acc[4] = {{}, {}, {}, {}};
        for (int k = 0; k < 64; k += 4) {
            const v2f a  = H2[abase + (k >> 1)];
            const int qb = (k >> 2) * 1024 + wbase;   // ((k>>2)*2+half)*512 + col
#pragma unroll
            for (int i = 0; i < 4; ++i) {
                const v2f b = w4p[qb + i * 16];
                acc[i] = wmma4(a, b, acc[i]);
            }
        }
        const int j = g >> 1;                // which of w0..w3 (uniform per g)
#pragma unroll
        for (int i = 0; i < 4; ++i) {
            const int u  = ((g & 1) * 4 + i) * 16 + lane15;   // channel 0..127
            const int u3 = 3 * u;
#pragma unroll
            for (int r = 0; r < 8; ++r) {
                const float w = acc[i][r] * 0.125f;           // 1/sqrt(64)
                if (j == 0) {            // m0a[u] -> col u
                    const float ss = s_up[sidx[r] + u];
                    atomicAdd(&msg[midx[r] + u], c * ss * y0a[r] * w);
                } else if (j == 1) {     // m0b[u] -> col 128+u
                    const int vb = vidx[r] + u3;
                    const float dot = v_up[vb + 0] * y1xa[r] + v_up[vb + 1] * y1ya[r] +
                                      v_up[vb + 2] * y1za[r];
                    atomicAdd(&msg[midx[r] + 128 + u], c3 * dot * w);
                } else if (j == 2) {     // m1a[u,m] -> col 256+3u+m
                    const float ss = s_up[sidx[r] + u];
                    const float t  = c * ss * w;
                    const int mb = midx[r] + 256 + u3;
                    atomicAdd(&msg[mb + 0], t * y1xa[r]);
                    atomicAdd(&msg[mb + 1], t * y1ya[r]);
                    atomicAdd(&msg[mb + 2], t * y1za[r]);
                } else {                 // m1b[u,m] -> col 640+3u+m
                    const int vb = vidx[r] + u3;
                    const float t  = c * y0a[r] * w;
                    const int mb = midx[r] + 640 + u3;
                    atomicAdd(&msg[mb + 0], t * v_up[vb + 0]);
                    atomicAdd(&msg[mb + 1], t * v_up[vb + 1]);
                    atomicAdd(&msg[mb + 2], t * v_up[vb + 2]);
                }
            }
        }
    }
}

// ---------------------------------------------------------------------------
// out_s = ms(256) @ W_lin0 /16 ; out_v[n,k,m] = sum_u mv[n,u,m] W_lin1[u,k] /16
// msg row: [0..255]=ms, [256 + 3u + m]=mv.  out row: [0..127]=out_s, [128+3k+m]=out_v. /10
__global__ void __launch_bounds__(256)
node_down_kernel(const float* __restrict__ msg, const v2f* __restrict__ L0p,
                 const v2f* __restrict__ L1p, float* __restrict__ out) {
    const int lane   = threadIdx.x & 31;
    const int wv     = threadIdx.x >> 5;
    const int lane15 = lane & 15;
    const int half   = lane >> 4;
    const int koff   = half * 2;
    const int mbase  = half * 8;
    const int n0     = blockIdx.x * 16;
    const int col    = wv * 16 + lane15;
    const int ab     = (n0 + lane15) * 1024;       // A row base (float units)
    const int ab2    = (n0 + lane15) * 512 + half; // A row base (v2f units)
    const int wbase  = half * 128 + col;           // B base (v2f units)

    const v2f* msg2 = (const v2f*)msg;

    v8f aS = {}, aV0 = {}, aV1 = {}, aV2 = {};
    for (int k = 0; k < 256; k += 4) {
        const int kk = k + koff;
        const int qb = (k >> 2) * 256 + wbase;
        const v2f b0 = L0p[qb];
        const v2f b1 = L1p[qb];
        const v2f a  = msg2[ab2 + (k >> 1)];
        aS = wmma4(a, b0, aS);
        const int vb = ab + 256 + kk * 3;
        v2f av;
        av.x = msg[vb + 0]; av.y = msg[vb + 3];
        aV0 = wmma4(av, b1, aV0);
        av.x = msg[vb + 1]; av.y = msg[vb + 4];
        aV1 = wmma4(av, b1, aV1);
        av.x = msg[vb + 2]; av.y = msg[vb + 5];
        aV2 = wmma4(av, b1, aV2);
    }

    const float sc = (1.0f / 16.0f) * 0.1f;  // 1/sqrt(256) * 1/AVG_NEIGH
#pragma unroll
    for (int r = 0; r < 8; ++r) {
        const int n = n0 + mbase + r;
        out[n * 512 + col] = aS[r] * sc;
        const int oo = n * 512 + 128 + col * 3;
        out[oo + 0] = aV0[r] * sc;
        out[oo + 1] = aV1[r] * sc;
        out[oo + 2] = aV2[r] * sc;
    }
}

// ---------------------------------------------------------------------------
extern "C" void kernel_launch(void* const* d_in, const int* in_sizes, int n_in,
                              void* d_out, int out_size, void* d_ws, size_t ws_size,
                              hipStream_t stream) {
    (void)in_sizes; (void)n_in; (void)out_size; (void)ws_size;
    const float* nf   = (const float*)d_in[0];
    const float* ea   = (const float*)d_in[1];
    const float* ef   = (const float*)d_in[2];
    const int*   snd  = (const int*)d_in[3];
    const int*   rcv  = (const int*)d_in[4];
    const float* Wup0 = (const float*)d_in[5];
    const float* Wup1 = (const float*)d_in[6];
    const float* w1   = (const float*)d_in[7];
    const float* w2   = (const float*)d_in[8];
    const float* w3   = (const float*)d_in[9];
    const float* w4   = (const float*)d_in[10];
    const float* Wl0  = (const float*)d_in[11];
    const float* Wl1  = (const float*)d_in[12];
    float* out = (float*)d_out;

    float* s_up = (float*)d_ws;                       // 20000*128  f32 (10.2 MB)
    float* v_up = s_up + (size_t)N_NODES * 128;       // 20000*384  f32 (30.7 MB)
    float* msgb = v_up + (size_t)N_NODES * 384;       // 20000*1024 f32 (81.9 MB)
    float* wp   = msgb + (size_t)N_NODES * 1024;      // packed weights (~0.56 MB)
    float* Wup0p = wp;  wp += 128 * 128;
    float* Wup1p = wp;  wp += 128 * 128;
    float* w1p   = wp;  wp += 8 * 64;
    float* w2p   = wp;  wp += 64 * 64;
    float* w3p   = wp;  wp += 64 * 64;
    float* w4p   = wp;  wp += 64 * 512;
    float* Wl0p  = wp;  wp += 256 * 128;
    float* Wl1p  = wp;  wp += 256 * 128;

    // zero the segment-sum accumulator (stays L2-resident: 82 MB < 192 MB L2)
    zero_kernel<<<2048, 256, 0, stream>>>((float4*)msgb, N_NODES * 1024 / 4);

    // pre-pack all weight matrices into WMMA-B fragment order
    pack_b_kernel<<<32, 256, 0, stream>>>(Wup0, (float2*)Wup0p, 128, 128);
    pack_b_kernel<<<32, 256, 0, stream>>>(Wup1, (float2*)Wup1p, 128, 128);
    pack_b_kernel<<<1,  256, 0, stream>>>(w1,   (float2*)w1p,   8,   64);
    pack_b_kernel<<<8,  256, 0, stream>>>(w2,   (float2*)w2p,   64,  64);
    pack_b_kernel<<<8,  256, 0, stream>>>(w3,   (float2*)w3p,   64,  64);
    pack_b_kernel<<<64, 256, 0, stream>>>(w4,   (float2*)w4p,   64,  512);
    pack_b_kernel<<<64, 256, 0, stream>>>(Wl0,  (float2*)Wl0p,  256, 128);
    pack_b_kernel<<<64, 256, 0, stream>>>(Wl1,  (float2*)Wl1p,  256, 128);

    node_up_kernel<<<N_NODES / 16, 256, 0, stream>>>(nf, (const v2f*)Wup0p,
                                                     (const v2f*)Wup1p, s_up, v_up);
    edge_kernel<<<N_EDGES / 64, 128, 0, stream>>>(ea, ef, snd, rcv,
                                                  (const v2f*)w1p, (const v2f*)w2p,
                                                  (const v2f*)w3p, (const v2f*)w4p,
                                                  s_up, v_up, msgb);
    node_down_kernel<<<N_NODES / 16, 256, 0, stream>>>(msgb, (const v2f*)Wl0p,
                                                       (const v2f*)Wl1p, out);
}